// SimpleGCN_37460704755814
// MI455X (gfx1250) — compile-verified
//
#include <hip/hip_runtime.h>
#include <math.h>

typedef float v2f __attribute__((ext_vector_type(2)));
typedef float v8f __attribute__((ext_vector_type(8)));

#define NODE_DIM 128
#define HIDDEN   64

// ---------------------------------------------------------------------------
// Stage 1: node_emb = relu(X @ W_enc + b_enc)    [N x 128] @ [128 x 64]
// One wave -> 16 rows x 64 cols (four 16x16 f32 WMMA accumulators), K step 4.
// W staged once per block into LDS, packed as K-pairs so the WMMA B-operand
// {W[k][n], W[k+1][n]} is a single aligned ds_load_b64.
// ---------------------------------------------------------------------------
__global__ __launch_bounds__(256) void gcn_encode(
    const float* __restrict__ X, const float* __restrict__ W,
    const float* __restrict__ bias, float* __restrict__ Y, int nrows) {
  __shared__ v2f sW[(NODE_DIM / 2) * HIDDEN];          // 4096 * 8B = 32 KB

  // Cooperative fill: entry kk*64+n = {W[2kk][n], W[2kk+1][n]} (coalesced rows)
  for (int idx = threadIdx.x; idx < (NODE_DIM / 2) * HIDDEN; idx += 256) {
    const int kk = idx >> 6;
    const int n  = idx & 63;
    v2f p = { W[(size_t)(2 * kk) * HIDDEN + n],
              W[(size_t)(2 * kk + 1) * HIDDEN + n] };
    sW[idx] = p;
  }
  __syncthreads();

  const int lane   = threadIdx.x & 31;
  const int wave   = threadIdx.x >> 5;
  const int m_base = (blockIdx.x * 8 + wave) * 16;
  if (m_base >= nrows) return;                 // uniform wave exit, EXEC all-1s

  const int nlo   = lane & 15;                 // M for A-op, N for B/C-op
  const int khalf = (lane >> 4) << 1;          // K sub-offset {0,2}

  v8f acc0 = {}, acc1 = {}, acc2 = {}, acc3 = {};
  const float* arow = X + (size_t)(m_base + nlo) * NODE_DIM;

  for (int k = 0; k < NODE_DIM; k += 4) {
    v2f a = *(const v2f*)(arow + k + khalf);   // A[m][k+khalf .. +1], 8B aligned
    const v2f* bp = &sW[((k + khalf) >> 1) * HIDDEN + nlo];
    v2f b0 = bp[0];
    v2f b1 = bp[16];
    v2f b2 = bp[32];
    v2f b3 = bp[48];
    acc0 = __builtin_amdgcn_wmma_f32_16x16x4_f32(false, a, false, b0, (short)0, acc0, false, false);
    acc1 = __builtin_amdgcn_wmma_f32_16x16x4_f32(false, a, false, b1, (short)0, acc1, false, false);
    acc2 = __builtin_amdgcn_wmma_f32_16x16x4_f32(false, a, false, b2, (short)0, acc2, false, false);
    acc3 = __builtin_amdgcn_wmma_f32_16x16x4_f32(false, a, false, b3, (short)0, acc3, false, false);
  }

  const int row_off = (lane >> 4) << 3;        // lanes 16-31 hold M = r+8
  #pragma unroll
  for (int r = 0; r < 8; ++r) {
    float* yr = Y + (size_t)(m_base + r + row_off) * HIDDEN;
    float v;
    v = acc0[r] + bias[nlo +  0]; yr[nlo +  0] = v > 0.f ? v : 0.f;
    v = acc1[r] + bias[nlo + 16]; yr[nlo + 16] = v > 0.f ? v : 0.f;
    v = acc2[r] + bias[nlo + 32]; yr[nlo + 32] = v > 0.f ? v : 0.f;
    v = acc3[r] + bias[nlo + 48]; yr[nlo + 48] = v > 0.f ? v : 0.f;
  }
}

// ---------------------------------------------------------------------------
// Stage 2: messages[dst] += node_emb[src]   (segment_sum over edges)
// One wave per edge; lane moves float2 (coalesced 256B row); f32 HW atomics.
// ---------------------------------------------------------------------------
__global__ __launch_bounds__(256) void gcn_scatter(
    const int* __restrict__ src, const int* __restrict__ dst,
    const float* __restrict__ emb, float* __restrict__ msgs, int nedges) {
  const int e = blockIdx.x * 8 + (threadIdx.x >> 5);
  if (e >= nedges) return;
  const int lane = threadIdx.x & 31;
  const int s = src[e];
  const int d = dst[e];
  const float2 v = ((const float2*)(emb + (size_t)s * HIDDEN))[lane];
  float* drow = msgs + (size_t)d * HIDDEN + lane * 2;
  unsafeAtomicAdd(drow,     v.x);   // global_atomic_add_f32
  unsafeAtomicAdd(drow + 1, v.y);
}

// ---------------------------------------------------------------------------
// Stage 3: emb2 = relu((node_emb + messages) @ W_conv + b_conv)
// Same WMMA structure, K=64; W_conv staged in LDS; writes in place over msgs.
// ---------------------------------------------------------------------------
__global__ __launch_bounds__(256) void gcn_conv(
    const float* __restrict__ emb, const float* __restrict__ msgs,
    const float* __restrict__ W, const float* __restrict__ bias,
    float* __restrict__ Y, int nrows) {
  __shared__ v2f sW[(HIDDEN / 2) * HIDDEN];            // 2048 * 8B = 16 KB

  for (int idx = threadIdx.x; idx < (HIDDEN / 2) * HIDDEN; idx += 256) {
    const int kk = idx >> 6;
    const int n  = idx & 63;
    v2f p = { W[(size_t)(2 * kk) * HIDDEN + n],
              W[(size_t)(2 * kk + 1) * HIDDEN + n] };
    sW[idx] = p;
  }
  __syncthreads();

  const int lane   = threadIdx.x & 31;
  const int wave   = threadIdx.x >> 5;
  const int m_base = (blockIdx.x * 8 + wave) * 16;
  if (m_base >= nrows) return;

  const int nlo   = lane & 15;
  const int khalf = (lane >> 4) << 1;

  v8f acc0 = {}, acc1 = {}, acc2 = {}, acc3 = {};
  const float* erow = emb  + (size_t)(m_base + nlo) * HIDDEN;
  const float* mrow = msgs + (size_t)(m_base + nlo) * HIDDEN;

  for (int k = 0; k < HIDDEN; k += 4) {
    v2f ea = *(const v2f*)(erow + k + khalf);
    v2f ma = *(const v2f*)(mrow + k + khalf);
    v2f a  = ea + ma;
    const v2f* bp = &sW[((k + khalf) >> 1) * HIDDEN + nlo];
    v2f b0 = bp[0];
    v2f b1 = bp[16];
    v2f b2 = bp[32];
    v2f b3 = bp[48];
    acc0 = __builtin_amdgcn_wmma_f32_16x16x4_f32(false, a, false, b0, (short)0, acc0, false, false);
    acc1 = __builtin_amdgcn_wmma_f32_16x16x4_f32(false, a, false, b1, (short)0, acc1, false, false);
    acc2 = __builtin_amdgcn_wmma_f32_16x16x4_f32(false, a, false, b2, (short)0, acc2, false, false);
    acc3 = __builtin_amdgcn_wmma_f32_16x16x4_f32(false, a, false, b3, (short)0, acc3, false, false);
  }

  const int row_off = (lane >> 4) << 3;
  #pragma unroll
  for (int r = 0; r < 8; ++r) {
    float* yr = Y + (size_t)(m_base + r + row_off) * HIDDEN;
    float v;
    v = acc0[r] + bias[nlo +  0]; yr[nlo +  0] = v > 0.f ? v : 0.f;
    v = acc1[r] + bias[nlo + 16]; yr[nlo + 16] = v > 0.f ? v : 0.f;
    v = acc2[r] + bias[nlo + 32]; yr[nlo + 32] = v > 0.f ? v : 0.f;
    v = acc3[r] + bias[nlo + 48]; yr[nlo + 48] = v > 0.f ? v : 0.f;
  }
}

// ---------------------------------------------------------------------------
// Stage 4: out[p] = sigmoid(emb2[post_mask[p]] . W_out + b_out)
// One wave per post; float2 partials + wave32 xor-shuffle reduction.
// ---------------------------------------------------------------------------
__global__ __launch_bounds__(256) void gcn_out(
    const float* __restrict__ emb2, const int* __restrict__ mask,
    const float* __restrict__ Wout, const float* __restrict__ bout,
    float* __restrict__ out, int nposts) {
  const int p = blockIdx.x * 8 + (threadIdx.x >> 5);
  if (p >= nposts) return;
  const int lane = threadIdx.x & 31;
  const int node = mask[p];
  const float2 a = ((const float2*)(emb2 + (size_t)node * HIDDEN))[lane];
  const float2 w = ((const float2*)Wout)[lane];
  float partial = a.x * w.x + a.y * w.y;
  #pragma unroll
  for (int off = 16; off > 0; off >>= 1)
    partial += __shfl_xor(partial, off, 32);
  if (lane == 0) {
    float x = partial + bout[0];
    out[p] = 1.0f / (1.0f + expf(-x));
  }
}

// ---------------------------------------------------------------------------
extern "C" void kernel_launch(void* const* d_in, const int* in_sizes, int n_in,
                              void* d_out, int out_size, void* d_ws, size_t ws_size,
                              hipStream_t stream) {
  const float* X     = (const float*)d_in[0];   // node_features [N,128]
  const int*   eidx  = (const int*)  d_in[1];   // edge_index [2,E] row-major
  const int*   pmask = (const int*)  d_in[2];   // post_mask [P]
  const float* Wenc  = (const float*)d_in[3];
  const float* benc  = (const float*)d_in[4];
  const float* Wconv = (const float*)d_in[5];
  const float* bconv = (const float*)d_in[6];
  const float* Wout  = (const float*)d_in[7];
  const float* bout  = (const float*)d_in[8];

  const int N = in_sizes[0] / NODE_DIM;
  const int E = in_sizes[1] / 2;
  const int P = in_sizes[2];
  const int* src = eidx;        // edge_index[0]
  const int* dst = eidx + E;    // edge_index[1]

  float* emb  = (float*)d_ws;                        // [N,64]  25.6 MB
  float* msgs = emb + (size_t)N * HIDDEN;            // [N,64]  25.6 MB (reused as emb2)

  const int tiles       = (N + 15) / 16;
  const int gemm_blocks = (tiles + 7) / 8;           // 8 waves (16-row slabs) per block

  gcn_encode<<<gemm_blocks, 256, 0, stream>>>(X, Wenc, benc, emb, N);
  hipMemsetAsync(msgs, 0, (size_t)N * HIDDEN * sizeof(float), stream);
  gcn_scatter<<<(E + 7) / 8, 256, 0, stream>>>(src, dst, emb, msgs, E);
  gcn_conv<<<gemm_blocks, 256, 0, stream>>>(emb, msgs, Wconv, bconv, msgs, N);
  gcn_out<<<(P + 7) / 8, 256, 0, stream>>>(msgs, pmask, Wout, bout, (float*)d_out, P);
}